// ManifoldHyperConnection_69312182223462
// MI455X (gfx1250) — compile-verified
//
#include <hip/hip_runtime.h>
#include <hip/hip_bf16.h>

// Fused ManifoldHyperConnection for gfx1250 (MI455X).
// HBM-bound kernel (512MB min traffic ~ 22us @ 23.3TB/s). The 24x4096
// projection GEMM (most of the FLOPs) runs on V_WMMA_F32_16X16X4_F32:
// 16 tokens/WG (M), 32 cols (N, 24 padded), K=4096 split over 8 waves.
// norm_w folded into weights (prep kernel -> d_ws), 1/rms deferred past
// the GEMM. Output pass applies per-token 4x4 G = Cayley(Q) + hpost*hpre^T.

typedef __attribute__((ext_vector_type(2))) float v2f;
typedef __attribute__((ext_vector_type(8))) float v8f;

#define D_DIM 4096
#define C_DIM 1024
#define NPROJ 24
#define KSTEPS 1024            // D_DIM / 4
#define WAVES 8
#define KPW (KSTEPS / WAVES)   // 128 K-steps per wave

__device__ __forceinline__ float sigmoidf_(float z) {
  return 1.f / (1.f + __expf(-z));
}

// ---------------------------------------------------------------------------
// Prep: build B-operand table wb[tile(2)][s(1024)][lane(32)] as float2 =
// (W'[col][4s+koff], W'[col][4s+1+koff]), col = tile*16 + lane%16 (0 pad >=24),
// koff = 2*(lane/16), W' = W * norm_w. Matches WMMA f32 16x16x4 B layout.
// ---------------------------------------------------------------------------
__global__ __launch_bounds__(256) void prep_wb_kernel(
    const float* __restrict__ Wpre, const float* __restrict__ Wpost,
    const float* __restrict__ Wres, const float* __restrict__ normw,
    float* __restrict__ wb) {
  int idx = blockIdx.x * 256 + threadIdx.x;   // 0 .. 65535 float2 entries
  int lane = idx & 31;
  int s = (idx >> 5) & (KSTEPS - 1);
  int tile = idx >> 15;
  int col = tile * 16 + (lane & 15);
  int koff = (lane >> 4) << 1;
  int d = (s << 2) + koff;
  float v0 = 0.f, v1 = 0.f;
  if (col < NPROJ) {
    const float* row;
    if (col < 4)      row = Wpre + col * D_DIM;
    else if (col < 8) row = Wpost + (col - 4) * D_DIM;
    else              row = Wres + (col - 8) * D_DIM;
    v0 = row[d] * normw[d];
    v1 = row[d + 1] * normw[d + 1];
  }
  wb[2 * idx]     = v0;
  wb[2 * idx + 1] = v1;
}

// ---------------------------------------------------------------------------
// Main fused kernel: 1 workgroup = 16 tokens, 256 threads (8 waves).
// ---------------------------------------------------------------------------
__global__ __launch_bounds__(256) void hyperconn_main_kernel(
    const float* __restrict__ x, const float* __restrict__ wb,
    const float* __restrict__ bpre, const float* __restrict__ bpost,
    const float* __restrict__ bres,
    const float* __restrict__ apre, const float* __restrict__ apost,
    const float* __restrict__ ares,
    float* __restrict__ out) {
  __shared__ float sspart[256];          // per-lane sum-of-squares partials
  __shared__ float red[WAVES * 2 * 8 * 32];  // per-wave WMMA accumulators
  __shared__ float Pmat[16 * 32];        // P[token][proj] (raw, pre-rinv)
  __shared__ float Gmat[16 * 16];        // per-token 4x4 mixing matrix
  __shared__ float rinvs[16];

  const int tid  = threadIdx.x;
  const int lane = tid & 31;
  const int w    = tid >> 5;
  const long t0  = (long)blockIdx.x * 16;

  // ---- Pass 1: WMMA projection P = x_tile @ W'^T, piggyback sum(x^2) ----
  // A layout (f32 16x16x4): lane L<16 -> row L, K=0..1 ; lane L>=16 -> row
  // L-16, K=2..3. Each wave owns a contiguous K range of 128 steps.
  const int rowA = lane & 15;
  const int koff = (lane >> 4) << 1;
  const float* xrow = x + (t0 + rowA) * (long)D_DIM + koff;
  const float* wb0 = wb + 2 * lane;                 // tile 0
  const float* wb1 = wb + 2 * (KSTEPS * 32) + 2 * lane;  // tile 1

  v8f c0 = {};
  v8f c1 = {};
  float ssq = 0.f;
  const int s0 = w * KPW;
  #pragma unroll 4
  for (int s = s0; s < s0 + KPW; ++s) {
    v2f a  = *(const v2f*)(xrow + 4 * s);
    v2f b0 = *(const v2f*)(wb0 + 64 * s);
    v2f b1 = *(const v2f*)(wb1 + 64 * s);
    ssq += a.x * a.x + a.y * a.y;
    c0 = __builtin_amdgcn_wmma_f32_16x16x4_f32(false, a, false, b0,
                                               (short)0, c0, false, false);
    c1 = __builtin_amdgcn_wmma_f32_16x16x4_f32(false, a, false, b1,
                                               (short)0, c1, false, false);
  }
  sspart[tid] = ssq;
  #pragma unroll
  for (int k = 0; k < 8; ++k) {
    red[((w * 2 + 0) * 8 + k) * 32 + lane] = c0[k];
    red[((w * 2 + 1) * 8 + k) * 32 + lane] = c1[k];
  }
  __syncthreads();

  // ---- rinv per token (each d covered exactly once across waves/lanes) ----
  if (tid < 16) {
    float t = 0.f;
    #pragma unroll
    for (int ww = 0; ww < WAVES; ++ww)
      t += sspart[ww * 32 + tid] + sspart[ww * 32 + 16 + tid];
    rinvs[tid] = rsqrtf(t * (1.0f / (float)D_DIM) + 1e-6f);
  }
  // ---- reduce the 8 partial accumulators into Pmat[token][col] ----
  for (int e = tid; e < 512; e += 256) {
    int tile = e >> 8;
    int k    = (e >> 5) & 7;
    int ln   = e & 31;
    float acc = 0.f;
    #pragma unroll
    for (int ww = 0; ww < WAVES; ++ww)
      acc += red[((ww * 2 + tile) * 8 + k) * 32 + ln];
    int Mrow = ((ln >> 4) << 3) + k;       // C/D layout: lanes>=16 hold M=8..15
    int Ncol = (tile << 4) + (ln & 15);
    Pmat[Mrow * 32 + Ncol] = acc;
  }
  __syncthreads();

  // ---- Per-token head math: sigmoids, Cayley(4x4), G = Q + hpost*hpre^T ----
  if (tid < 16) {
    const float rinv = rinvs[tid];
    const float aP = apre[0], aO = apost[0], aR = ares[0];
    const float* Prow = &Pmat[tid * 32];
    float hpre[4], hpost[4];
    #pragma unroll
    for (int j = 0; j < 4; ++j)
      hpre[j] = sigmoidf_(aP * Prow[j] * rinv + bpre[j]);
    #pragma unroll
    for (int j = 0; j < 4; ++j)
      hpost[j] = 2.f * sigmoidf_(aO * Prow[4 + j] * rinv + bpost[j]);
    float Mm[4][4];
    #pragma unroll
    for (int a4 = 0; a4 < 4; ++a4)
      #pragma unroll
      for (int b4 = 0; b4 < 4; ++b4)
        Mm[a4][b4] = aR * Prow[8 + a4 * 4 + b4] * rinv + bres[a4 * 4 + b4];
    // Cayley: solve (I - S) Q = (I + S), S = skew(M). I-S has SPD symmetric
    // part -> Gauss-Jordan without pivoting is safe.
    float A[4][4], Q[4][4];
    #pragma unroll
    for (int i = 0; i < 4; ++i)
      #pragma unroll
      for (int j = 0; j < 4; ++j) {
        float sk = 0.5f * (Mm[i][j] - Mm[j][i]);
        float id = (i == j) ? 1.f : 0.f;
        A[i][j] = id - sk;
        Q[i][j] = id + sk;
      }
    #pragma unroll
    for (int k = 0; k < 4; ++k) {
      float inv = 1.f / A[k][k];
      #pragma unroll
      for (int j = 0; j < 4; ++j) { A[k][j] *= inv; Q[k][j] *= inv; }
      #pragma unroll
      for (int i = 0; i < 4; ++i)
        if (i != k) {
          float f = A[i][k];
          #pragma unroll
          for (int j = 0; j < 4; ++j) {
            A[i][j] -= f * A[k][j];
            Q[i][j] -= f * Q[k][j];
          }
        }
    }
    #pragma unroll
    for (int m = 0; m < 4; ++m)
      #pragma unroll
      for (int nn = 0; nn < 4; ++nn)
        Gmat[tid * 16 + m * 4 + nn] = Q[m][nn] + hpost[m] * hpre[nn];
  }
  __syncthreads();

  // ---- Pass 2: out[m][c] = sum_n G[m][n] * x[n][c]; x re-read hits L2 ----
  {
    const int tok = tid >> 4;
    const int li  = tid & 15;
    const float* xin = x + (t0 + tok) * (long)D_DIM;
    float* xo = out + (t0 + tok) * (long)D_DIM;
    float G[16];
    #pragma unroll
    for (int i = 0; i < 16; ++i) G[i] = Gmat[tok * 16 + i];
    for (int j = 0; j < 16; ++j) {
      int c = (li + (j << 4)) << 2;   // float4-aligned column
      float4 x0 = *(const float4*)(xin + 0 * C_DIM + c);
      float4 x1 = *(const float4*)(xin + 1 * C_DIM + c);
      float4 x2 = *(const float4*)(xin + 2 * C_DIM + c);
      float4 x3 = *(const float4*)(xin + 3 * C_DIM + c);
      #pragma unroll
      for (int m = 0; m < 4; ++m) {
        float g0 = G[m * 4 + 0], g1 = G[m * 4 + 1];
        float g2 = G[m * 4 + 2], g3 = G[m * 4 + 3];
        float4 o4;
        o4.x = g0 * x0.x + g1 * x1.x + g2 * x2.x + g3 * x3.x;
        o4.y = g0 * x0.y + g1 * x1.y + g2 * x2.y + g3 * x3.y;
        o4.z = g0 * x0.z + g1 * x1.z + g2 * x2.z + g3 * x3.z;
        o4.w = g0 * x0.w + g1 * x1.w + g2 * x2.w + g3 * x3.w;
        *(float4*)(xo + m * C_DIM + c) = o4;
      }
    }
  }
}

extern "C" void kernel_launch(void* const* d_in, const int* in_sizes, int n_in,
                              void* d_out, int out_size, void* d_ws, size_t ws_size,
                              hipStream_t stream) {
  const float* x     = (const float*)d_in[0];
  const float* normw = (const float*)d_in[1];
  const float* Wpre  = (const float*)d_in[2];
  const float* Wpost = (const float*)d_in[3];
  const float* Wres  = (const float*)d_in[4];
  const float* bpre  = (const float*)d_in[5];
  const float* bpost = (const float*)d_in[6];
  const float* bres  = (const float*)d_in[7];
  const float* apre  = (const float*)d_in[8];
  const float* apost = (const float*)d_in[9];
  const float* ares  = (const float*)d_in[10];
  float* out = (float*)d_out;
  float* wb  = (float*)d_ws;   // needs 2*1024*32*2*4 = 512 KB scratch

  const int BL = in_sizes[0] / D_DIM;   // 16384 tokens

  // 65536 float2 entries -> 256 blocks x 256 threads
  hipLaunchKernelGGL(prep_wb_kernel, dim3(256), dim3(256), 0, stream,
                     Wpre, Wpost, Wres, normw, wb);
  hipLaunchKernelGGL(hyperconn_main_kernel, dim3(BL / 16), dim3(256), 0, stream,
                     x, wb, bpre, bpost, bres, apre, apost, ares, out);
  (void)n_in; (void)out_size; (void)ws_size;
}